// ShapeEncoderDGCNN_46024869544177
// MI455X (gfx1250) — compile-verified
//
#include <hip/hip_runtime.h>
#include <hip/hip_bf16.h>
#include <cstdint>

typedef float v2f __attribute__((ext_vector_type(2)));
typedef float v8f __attribute__((ext_vector_type(8)));

#define NPTS 4096
#define NB 4
#define KNN 20
#define EPSV 1e-5f
#define SLOPE 0.2f

__device__ __forceinline__ float lrelu(float v){ return v >= 0.f ? v : SLOPE * v; }

// ---------- transpose (B,N,3) -> (B,3,N) ----------
__global__ void k_transpose3(const float* __restrict__ x, float* __restrict__ xt){
  int i = blockIdx.x*blockDim.x + threadIdx.x;   // over B*3*N
  int n = i % NPTS; int t = i / NPTS; int c = t % 3; int b = t / 3;
  xt[i] = x[((size_t)b*NPTS + n)*3 + c];
}

// ---------- norms[b][n] = sum_c F[b][c][n]^2 ----------
__global__ void k_norms(const float* __restrict__ F, int C, size_t fbs,
                        float* __restrict__ nrm){
  int i = blockIdx.x*blockDim.x + threadIdx.x;   // over B*N
  int n = i % NPTS; int b = i / NPTS;
  const float* f = F + (size_t)b*fbs;
  float s = 0.f;
  for(int c = 0; c < C; ++c){ float v = f[(size_t)c*NPTS + n]; s += v*v; }
  nrm[i] = s;
}

// ---------- knn: per 64-row tile, WMMA gram + per-thread top-20 ----------
__global__ void __launch_bounds__(64) k_knn(const float* __restrict__ F, int Creal, int Cp,
                     size_t fbs, const float* __restrict__ nrm, int* __restrict__ idxout){
  __shared__ float As[64*64];   // [row][k], stride Cp
  __shared__ float Bs[64*64];   // [k][col], stride 64
  __shared__ float Ds[64*64];   // 64x64 dot tile
  const int b = blockIdx.y;
  const int row0 = blockIdx.x*64;
  const int tid = threadIdx.x;
  const int wv = tid>>5, lane = tid&31, half = lane>>4, l16 = lane&15;
  const float* Fb = F + (size_t)b*fbs;

  for(int k = 0; k < Cp; ++k)
    As[tid*Cp + k] = (k < Creal) ? Fb[(size_t)k*NPTS + row0 + tid] : 0.f;

  const float rn = nrm[b*NPTS + row0 + tid];
  float bd[KNN]; int bi[KNN];
  for(int j = 0; j < KNN; ++j){ bd[j] = -3.0e38f; bi[j] = 0; }
  __syncthreads();

  for(int ct = 0; ct < NPTS/64; ++ct){
    const int c0 = ct*64;
    for(int k = 0; k < Cp; ++k)
      Bs[k*64 + tid] = (k < Creal) ? Fb[(size_t)k*NPTS + c0 + tid] : 0.f;
    __syncthreads();

    for(int s = wv; s < 16; s += 2){
      const int sr = s >> 2, sc = s & 3;
      v8f acc = {0.f,0.f,0.f,0.f,0.f,0.f,0.f,0.f};
      const int m = sr*16 + l16;
      for(int k0 = 0; k0 < Cp; k0 += 4){
        v2f a, bb;
        a.x = As[m*Cp + k0 + 2*half + 0];
        a.y = As[m*Cp + k0 + 2*half + 1];
        bb.x = Bs[(k0 + half    )*64 + sc*16 + l16];
        bb.y = Bs[(k0 + 2 + half)*64 + sc*16 + l16];
        acc = __builtin_amdgcn_wmma_f32_16x16x4_f32(false, a, false, bb,
                                                    (short)0, acc, false, false);
      }
      #pragma unroll
      for(int v = 0; v < 8; ++v)
        Ds[(sr*16 + v + 8*half)*64 + sc*16 + l16] = acc[v];
    }
    __syncthreads();

    for(int c = 0; c < 64; ++c){
      float pd = 2.f*Ds[tid*64 + c] - rn - nrm[b*NPTS + c0 + c];
      if(pd > bd[KNN-1]){
        int p = KNN-1;
        while(p > 0 && pd > bd[p-1]){ bd[p] = bd[p-1]; bi[p] = bi[p-1]; --p; }
        bd[p] = pd; bi[p] = c0 + c;
      }
    }
    __syncthreads();
  }
  int* op = idxout + ((size_t)b*NPTS + row0 + tid)*KNN;
  for(int j = 0; j < KNN; ++j) op[j] = bi[j];
}

// ---------- build stacked edge weight: rows 0-63 = w_a, 64-127 = w_b - w_a ----------
__global__ void k_prepw(const float* __restrict__ w, int C, int Cp, float* __restrict__ Wd){
  int i = blockIdx.x*blockDim.x + threadIdx.x;
  if(i >= 128*Cp) return;
  int r = i / Cp, k = i % Cp;
  float v = 0.f;
  if(k < C) v = (r < 64) ? w[r*2*C + k]
                         : (w[(r-64)*2*C + C + k] - w[(r-64)*2*C + k]);
  Wd[i] = v;
}

// ---------- generic WMMA GEMM: OUT(b,MO,N) = W(MO x C) @ F(b,C,N) ----------
__global__ void __launch_bounds__(128) k_wgemm(const float* __restrict__ W, int ldw,
                     int Creal, int Cpad, const float* __restrict__ F, size_t fbs,
                     float* __restrict__ OUT, size_t obs){
  __shared__ float Ws[64*64];   // [mr][kk]
  __shared__ float Fs[64*64];   // [kk][nc]
  const int b  = blockIdx.z;
  const int n0 = blockIdx.x*64;
  const int m0 = blockIdx.y*64;
  const int tid = threadIdx.x;
  const int wv = tid>>5, lane = tid&31, half = lane>>4, l16 = lane&15;
  const float* Fb = F + (size_t)b*fbs;

  v8f acc[4];
  #pragma unroll
  for(int sc = 0; sc < 4; ++sc){ v8f z = {0.f,0.f,0.f,0.f,0.f,0.f,0.f,0.f}; acc[sc] = z; }

  for(int kc = 0; kc < Cpad; kc += 64){
    for(int i = tid; i < 4096; i += 128){
      int mr = i >> 6, kk = i & 63;
      Ws[i] = (kc + kk < Cpad) ? W[(size_t)(m0 + mr)*ldw + kc + kk] : 0.f;
    }
    for(int i = tid; i < 4096; i += 128){
      int kk = i >> 6, nc = i & 63;
      Fs[i] = (kc + kk < Creal) ? Fb[(size_t)(kc + kk)*NPTS + n0 + nc] : 0.f;
    }
    __syncthreads();
    int KL = Cpad - kc; if(KL > 64) KL = 64;
    for(int k0 = 0; k0 < KL; k0 += 4){
      v2f a;
      a.x = Ws[(wv*16 + l16)*64 + k0 + 2*half + 0];
      a.y = Ws[(wv*16 + l16)*64 + k0 + 2*half + 1];
      #pragma unroll
      for(int sc = 0; sc < 4; ++sc){
        v2f bb;
        bb.x = Fs[(k0 + half    )*64 + sc*16 + l16];
        bb.y = Fs[(k0 + 2 + half)*64 + sc*16 + l16];
        acc[sc] = __builtin_amdgcn_wmma_f32_16x16x4_f32(false, a, false, bb,
                                                        (short)0, acc[sc], false, false);
      }
    }
    __syncthreads();
  }
  #pragma unroll
  for(int sc = 0; sc < 4; ++sc){
    #pragma unroll
    for(int v = 0; v < 8; ++v){
      int m = m0 + wv*16 + v + 8*half;
      OUT[(size_t)b*obs + (size_t)m*NPTS + n0 + sc*16 + l16] = acc[sc][v];
    }
  }
}

// ---------- gather neighbors, max over k, per-block BN partial sums ----------
__global__ void __launch_bounds__(256) k_edge_reduce(const float* __restrict__ GB,
                     const int* __restrict__ idx, float* __restrict__ Mb,
                     float* __restrict__ epart){
  __shared__ int   sidx[64*KNN];
  __shared__ float ps[256], pq[256];
  const int b = blockIdx.y, n0 = blockIdx.x*64, tid = threadIdx.x;
  for(int i = tid; i < 64*KNN; i += 256)
    sidx[i] = idx[((size_t)b*NPTS + n0)*KNN + i];
  __syncthreads();
  const int o = tid & 63;
  const float* G  = GB + ((size_t)b*128 + o     )*NPTS;
  const float* Ba = GB + ((size_t)b*128 + 64 + o)*NPTS;
  float s = 0.f, q = 0.f;
  for(int t = tid >> 6; t < 64; t += 4){
    const int n = n0 + t;
    const float base = Ba[n];
    float mx = -3.0e38f;
    for(int k = 0; k < KNN; ++k){
      float v = G[sidx[t*KNN + k]] + base;
      mx = fmaxf(mx, v); s += v; q += v*v;
    }
    Mb[((size_t)b*64 + o)*NPTS + n] = mx;
  }
  ps[tid] = s; pq[tid] = q;
  __syncthreads();
  if(tid < 64){
    float S = ps[tid] + ps[tid+64] + ps[tid+128] + ps[tid+192];
    float Q = pq[tid] + pq[tid+64] + pq[tid+128] + pq[tid+192];
    int blk = blockIdx.y*(NPTS/64) + blockIdx.x;   // 0..255
    epart[(size_t)blk*128 + tid]      = S;
    epart[(size_t)blk*128 + 64 + tid] = Q;
  }
}

__global__ void k_edge_stats(const float* __restrict__ epart, float* __restrict__ stats){
  int o = threadIdx.x;  // 64 threads
  float S = 0.f, Q = 0.f;
  for(int blk = 0; blk < 256; ++blk){
    S += epart[(size_t)blk*128 + o];
    Q += epart[(size_t)blk*128 + 64 + o];
  }
  stats[o] = S; stats[1024 + o] = Q;
}

// ---------- BN + lrelu for edge layer, write into X channel slice ----------
__global__ void k_bn_edge(const float* __restrict__ Mb, const float* __restrict__ stats,
                          const float* __restrict__ gamma, const float* __restrict__ beta,
                          float* __restrict__ X, int coff, float cnt){
  int i = blockIdx.x*blockDim.x + threadIdx.x;   // B*64*N
  int n = i % NPTS; int t = i / NPTS; int o = t & 63; int b = t >> 6;
  float mean = stats[o]/cnt;
  float var  = stats[1024+o]/cnt - mean*mean;
  float v = (Mb[i] - mean)*rsqrtf(var + EPSV)*gamma[o] + beta[o];
  X[((size_t)(b*256 + coff + o))*NPTS + n] = lrelu(v);
}

// ---------- layer-5 per-batch BN partial stats ----------
__global__ void k_stats5(const float* __restrict__ Hb, float* __restrict__ p5, int b){
  __shared__ float ss[256], sq[256];
  const int o = blockIdx.x, tid = threadIdx.x;
  float s = 0.f, q = 0.f;
  for(int n = tid; n < NPTS; n += 256){
    float v = Hb[(size_t)o*NPTS + n]; s += v; q += v*v;
  }
  ss[tid] = s; sq[tid] = q; __syncthreads();
  for(int st = 128; st > 0; st >>= 1){
    if(tid < st){ ss[tid] += ss[tid+st]; sq[tid] += sq[tid+st]; }
    __syncthreads();
  }
  if(tid == 0){ p5[(size_t)b*2048 + o] = ss[0]; p5[(size_t)b*2048 + 1024 + o] = sq[0]; }
}

__global__ void k_stats5fin(const float* __restrict__ p5, float* __restrict__ stats){
  int o = blockIdx.x*blockDim.x + threadIdx.x;   // 1024
  float S = 0.f, Q = 0.f;
  for(int b = 0; b < NB; ++b){
    S += p5[(size_t)b*2048 + o];
    Q += p5[(size_t)b*2048 + 1024 + o];
  }
  stats[o] = S; stats[1024 + o] = Q;
}

// ---------- layer-5 BN + lrelu + tile max/sum partials ----------
__global__ void k_apply5(const float* __restrict__ Hb, const float* __restrict__ stats,
                         const float* __restrict__ g5, const float* __restrict__ b5,
                         float* __restrict__ pmax, float* __restrict__ psum, int b){
  __shared__ float sm[256], sx[256];
  const int o = blockIdx.x, tid = threadIdx.x;
  const int n = blockIdx.y*256 + tid;
  const float cnt = (float)NB*(float)NPTS;
  float mean = stats[o]/cnt;
  float var  = stats[1024+o]/cnt - mean*mean;
  float v = (Hb[(size_t)o*NPTS + n] - mean)*rsqrtf(var + EPSV)*g5[o] + b5[o];
  v = lrelu(v);
  sm[tid] = v; sx[tid] = v; __syncthreads();
  for(int st = 128; st > 0; st >>= 1){
    if(tid < st){ sm[tid] = fmaxf(sm[tid], sm[tid+st]); sx[tid] += sx[tid+st]; }
    __syncthreads();
  }
  if(tid == 0){
    size_t base = ((size_t)b*1024 + o)*16 + blockIdx.y;
    pmax[base] = sm[0]; psum[base] = sx[0];
  }
}

__global__ void k_redpart(const float* __restrict__ pmax, const float* __restrict__ psum,
                          float* __restrict__ hma){
  int i = blockIdx.x*blockDim.x + threadIdx.x;   // B*1024
  int b = i >> 10, o = i & 1023;
  float mx = -3.0e38f, s = 0.f;
  for(int t = 0; t < 16; ++t){
    mx = fmaxf(mx, pmax[(size_t)i*16 + t]);
    s += psum[(size_t)i*16 + t];
  }
  hma[(size_t)b*2048 + o]        = mx;
  hma[(size_t)b*2048 + 1024 + o] = s / (float)NPTS;
}

// ---------- final FC: y[b][o] = sum_c g[b][c] * wl[o][c] ----------
__global__ void k_fc(const float* __restrict__ hma, const float* __restrict__ wl,
                     float* __restrict__ y){
  int i = blockIdx.x*blockDim.x + threadIdx.x;   // B*1024
  int b = i >> 10, o = i & 1023;
  const float* g  = hma + (size_t)b*2048;
  const float* wr = wl  + (size_t)o*2048;
  float s = 0.f;
  for(int c = 0; c < 2048; ++c) s += g[c]*wr[c];
  y[i] = s;
}

// ---------- final BN over batch + lrelu ----------
__global__ void k_final(const float* __restrict__ y, const float* __restrict__ g6,
                        const float* __restrict__ b6, float* __restrict__ out){
  int o = blockIdx.x*blockDim.x + threadIdx.x;   // 1024
  float v0 = y[o], v1 = y[1024+o], v2 = y[2048+o], v3 = y[3072+o];
  float m = 0.25f*(v0+v1+v2+v3);
  float d0 = v0-m, d1 = v1-m, d2 = v2-m, d3 = v3-m;
  float var = 0.25f*(d0*d0 + d1*d1 + d2*d2 + d3*d3);
  float sc = rsqrtf(var + EPSV)*g6[o];
  out[o]        = lrelu(d0*sc + b6[o]);
  out[1024 + o] = lrelu(d1*sc + b6[o]);
  out[2048 + o] = lrelu(d2*sc + b6[o]);
  out[3072 + o] = lrelu(d3*sc + b6[o]);
}

extern "C" void kernel_launch(void* const* d_in, const int* in_sizes, int n_in,
                              void* d_out, int out_size, void* d_ws, size_t ws_size,
                              hipStream_t stream) {
  const float* x  = (const float*)d_in[0];
  const float* wlist[4] = {(const float*)d_in[1], (const float*)d_in[4],
                           (const float*)d_in[7], (const float*)d_in[10]};
  const float* glist[4] = {(const float*)d_in[2], (const float*)d_in[5],
                           (const float*)d_in[8], (const float*)d_in[11]};
  const float* blist[4] = {(const float*)d_in[3], (const float*)d_in[6],
                           (const float*)d_in[9], (const float*)d_in[12]};
  const float* w5 = (const float*)d_in[13];
  const float* g5 = (const float*)d_in[14];
  const float* b5 = (const float*)d_in[15];
  const float* wl = (const float*)d_in[16];
  const float* g6 = (const float*)d_in[17];
  const float* b6 = (const float*)d_in[18];
  float* out = (float*)d_out;

  float* ws = (float*)d_ws;
  size_t p = 0;
  float* X     = ws + p; p += (size_t)NB*256*NPTS;        // (B,256,N) x1..x4
  float* xt0   = ws + p; p += (size_t)NB*3*NPTS;          // (B,3,N)
  float* nrm   = ws + p; p += (size_t)NB*NPTS;            // (B,N)
  int*   idx   = (int*)(ws + p); p += (size_t)NB*NPTS*KNN;
  float* GB    = ws + p; p += (size_t)NB*128*NPTS;        // G + Base
  float* Mb    = ws + p; p += (size_t)NB*64*NPTS;         // pre-BN max
  float* stats = ws + p; p += 2048;
  float* Wd    = ws + p; p += 128*64;
  float* epart = ws + p; p += 256*128;
  float* Hb    = ws + p; p += (size_t)1024*NPTS;          // per-batch layer5 out
  float* p5    = ws + p; p += (size_t)NB*2048;
  float* pmax  = ws + p; p += (size_t)NB*1024*16;
  float* psum  = ws + p; p += (size_t)NB*1024*16;
  float* hma   = ws + p; p += (size_t)NB*2048;
  float* y     = ws + p; p += (size_t)NB*1024;
  (void)ws_size; (void)n_in; (void)in_sizes; (void)out_size;

  k_transpose3<<<(NB*3*NPTS)/256, 256, 0, stream>>>(x, xt0);

  for(int L = 0; L < 4; ++L){
    const int C  = L ? 64 : 3;
    const int Cp = L ? 64 : 4;
    const float* F = L ? (X + (size_t)(L-1)*64*NPTS) : xt0;
    const size_t fbs = L ? (size_t)256*NPTS : (size_t)3*NPTS;

    k_norms<<<(NB*NPTS)/256, 256, 0, stream>>>(F, C, fbs, nrm);
    k_knn<<<dim3(NPTS/64, NB), 64, 0, stream>>>(F, C, Cp, fbs, nrm, idx);
    k_prepw<<<(128*Cp + 255)/256, 256, 0, stream>>>(wlist[L], C, Cp, Wd);
    k_wgemm<<<dim3(NPTS/64, 2, NB), 128, 0, stream>>>(Wd, Cp, C, Cp, F, fbs,
                                                      GB, (size_t)128*NPTS);
    k_edge_reduce<<<dim3(NPTS/64, NB), 256, 0, stream>>>(GB, idx, Mb, epart);
    k_edge_stats<<<1, 64, 0, stream>>>(epart, stats);
    k_bn_edge<<<(NB*64*NPTS)/256, 256, 0, stream>>>(Mb, stats, glist[L], blist[L],
                                                    X, L*64, (float)NB*NPTS*KNN);
  }

  // layer 5: two passes (stats, then apply) to avoid materializing all batches
  for(int b = 0; b < NB; ++b){
    k_wgemm<<<dim3(NPTS/64, 16, 1), 128, 0, stream>>>(w5, 256, 256, 256,
                                  X + (size_t)b*256*NPTS, 0, Hb, 0);
    k_stats5<<<1024, 256, 0, stream>>>(Hb, p5, b);
  }
  k_stats5fin<<<4, 256, 0, stream>>>(p5, stats);
  for(int b = 0; b < NB; ++b){
    k_wgemm<<<dim3(NPTS/64, 16, 1), 128, 0, stream>>>(w5, 256, 256, 256,
                                  X + (size_t)b*256*NPTS, 0, Hb, 0);
    k_apply5<<<dim3(1024, 16), 256, 0, stream>>>(Hb, stats, g5, b5, pmax, psum, b);
  }
  k_redpart<<<(NB*1024)/256, 256, 0, stream>>>(pmax, psum, hma);
  k_fc<<<(NB*1024)/256, 256, 0, stream>>>(hma, wl, y);
  k_final<<<1024/256, 256, 0, stream>>>(y, g6, b6, out);
}